// GCN_layer_60069412602159
// MI455X (gfx1250) — compile-verified
//
#include <hip/hip_runtime.h>

// GCN layer: symmetric-normalized SpMM done with V_WMMA_F32_16X16X4_F32.
// Memory-bound (feature gather is L2-resident: 51.2 MB table in 192 MB L2),
// so full-precision f32 WMMA is free and exact.

typedef __attribute__((ext_vector_type(2))) float v2f;
typedef __attribute__((ext_vector_type(8))) float v8f;

#define D_FEAT 128
#define TILE_E 128   // edges staged in LDS per iteration (multiple of 4)
#define EPS    1e-12f

// ---------------------------------------------------------------- utilities
__global__ void zero_f32_kernel(float* __restrict__ p, int n) {
    int i = blockIdx.x * blockDim.x + threadIdx.x;
    if (i < n) p[i] = 0.0f;
}

__global__ void degree_kernel(const int* __restrict__ erow,
                              const float* __restrict__ evals,
                              float* __restrict__ rowsum, int E) {
    int e = blockIdx.x * blockDim.x + threadIdx.x;
    if (e < E) atomicAdd(&rowsum[erow[e]], evals[e]);
}

__global__ void dinv_kernel(const float* __restrict__ rowsum,
                            float* __restrict__ dinv, int n) {
    int i = blockIdx.x * blockDim.x + threadIdx.x;
    if (i < n) dinv[i] = rsqrtf(fmaxf(rowsum[i], EPS));
}

__global__ void copy_feat_kernel(const float4* __restrict__ src,
                                 float4* __restrict__ dst, int n4) {
    int i = blockIdx.x * blockDim.x + threadIdx.x;
    if (i < n4) dst[i] = src[i];
}

// ---------------------------------------------------------------- SpMM
// One workgroup = 16 output rows. 8 waves; wave w owns feature cols [16w,16w+16).
// Per 4-edge chunk: A[16x4] carries (row-select * norm), B[4x16] carries
// gathered feature values; D += A*B is an exact fp32 segmented accumulate.
// LDS records are padded to a multiple of 4 edges with sentinels so the inner
// loop has NO lane-divergent control flow (EXEC stays all ones). The row
// selection uses a {1.0,0.0} mask MULTIPLY (not a select) so LLVM cannot sink
// the LDS load into a predicated (exec-masked) load.
__global__ __launch_bounds__(256)
void spmm_wmma_kernel(const float* __restrict__ feat,
                      const int*   __restrict__ erow,
                      const int*   __restrict__ ecol,
                      const float* __restrict__ evals,
                      const float* __restrict__ dinv,
                      const int*   __restrict__ index,
                      float*       __restrict__ out,
                      int N, int E) {
    // {rel_row, norm_bits} packed per edge; 16B-aligned so a lane's two
    // consecutive records (e0 even) load as one ds_load_b128.
    __shared__ __align__(16) int2 s_rn[TILE_E];
    __shared__ __align__(8)  int  s_col[TILE_E];

    const int r0     = blockIdx.x * 16;
    const int tid    = threadIdx.x;
    const int lane   = tid & 31;
    const int hi     = lane >> 4;        // 0: lanes 0-15 (K=0,1), 1: lanes 16-31 (K=2,3)
    const int m      = lane & 15;        // matrix row (A) / column (B,C)
    const int coloff = (tid >> 5) * 16;  // this wave's 16-column feature slice

    // lower_bound over sorted edge_row (uniform across the workgroup)
    auto lb = [&](int v) {
        int lo = 0, len = E;
        while (len > 0) {
            int half = len >> 1;
            int mid  = lo + half;
            if (erow[mid] < v) { lo = mid + 1; len -= half + 1; }
            else               { len = half; }
        }
        return lo;
    };
    const int start = lb(r0);
    const int end   = lb(r0 + 16);

    v8f acc = {0.f, 0.f, 0.f, 0.f, 0.f, 0.f, 0.f, 0.f};

    for (int t = start; t < end; t += TILE_E) {
        const int cnt  = min(TILE_E, end - t);
        const int cntp = (cnt + 3) & ~3;             // padded edge count
        __syncthreads();
        for (int i = tid; i < cnt; i += 256) {
            int e = t + i;
            int r = erow[e];
            int c = ecol[e];
            float nv = evals[e] * dinv[r] * dinv[c]; // D^-1/2 A D^-1/2
            s_rn[i]  = make_int2(r - r0, __float_as_int(nv));
            s_col[i] = c;
        }
        // sentinel pad: rel_row=-1 never matches m, norm=0, col=0 (safe addr)
        for (int i = cnt + tid; i < cntp; i += 256) {
            s_rn[i]  = make_int2(-1, 0);
            s_col[i] = 0;
        }
        __syncthreads();

        for (int k = 0; k < cntp; k += 4) {          // uniform trip count
            const int e0 = k + 2 * hi;               // even -> aligned vector loads
            const int4 rn = *reinterpret_cast<const int4*>(&s_rn[e0]);
            const int2 cc = *reinterpret_cast<const int2*>(&s_col[e0]);

            // {1.0, 0.0} masks: multiply keeps the record load unconditional
            // (no x*0==0 folding without fast-math -> no exec-masked load).
            const float w0 = (rn.x == m) ? 1.0f : 0.0f;
            const float w1 = (rn.z == m) ? 1.0f : 0.0f;

            v2f a, b;
            a.x = w0 * __int_as_float(rn.y);                 // A[m][K=2*hi+0]
            a.y = w1 * __int_as_float(rn.w);                 // A[m][K=2*hi+1]
            b.x = feat[cc.x * D_FEAT + coloff + m];          // B[K=2*hi+0][m]
            b.y = feat[cc.y * D_FEAT + coloff + m];          // B[K=2*hi+1][m]

            acc = __builtin_amdgcn_wmma_f32_16x16x4_f32(
                /*neg_a=*/false, a, /*neg_b=*/false, b,
                /*c_mod=*/(short)0, acc, /*reuse_a=*/false, /*reuse_b=*/false);
        }
    }

    // C/D layout: VGPR r -> output row r + 8*hi, column m. Scatter through index.
    #pragma unroll
    for (int r = 0; r < 8; ++r) {
        const int grow = r0 + r + 8 * hi;
        if (grow < N) {
            const int drow = index[grow];
            out[drow * D_FEAT + coloff + m] = acc[r];
        }
    }
}

// ---------------------------------------------------------------- launch
extern "C" void kernel_launch(void* const* d_in, const int* in_sizes, int n_in,
                              void* d_out, int out_size, void* d_ws, size_t ws_size,
                              hipStream_t stream) {
    const float* feat  = (const float*)d_in[0];  // features  [N, 128]
    const int*   erow  = (const int*)  d_in[1];  // edge_row  [E] (sorted)
    const int*   ecol  = (const int*)  d_in[2];  // edge_col  [E]
    const float* evals = (const float*)d_in[3];  // edge_vals [E]
    const int*   index = (const int*)  d_in[4];  // index     [N]

    const int N = in_sizes[0] / D_FEAT;
    const int E = in_sizes[1];

    float* out    = (float*)d_out;
    float* rowsum = (float*)d_ws;        // N floats
    float* dinv   = rowsum + N;          // N floats

    zero_f32_kernel<<<(N + 255) / 256, 256, 0, stream>>>(rowsum, N);
    degree_kernel  <<<(E + 255) / 256, 256, 0, stream>>>(erow, evals, rowsum, E);
    dinv_kernel    <<<(N + 255) / 256, 256, 0, stream>>>(rowsum, dinv, N);

    const int n4 = (N * D_FEAT) / 4;
    copy_feat_kernel<<<(n4 + 255) / 256, 256, 0, stream>>>(
        (const float4*)feat, (float4*)out, n4);

    const int nblocks = (N + 15) / 16;
    spmm_wmma_kernel<<<nblocks, 256, 0, stream>>>(
        feat, erow, ecol, evals, dinv, index, out, N, E);
}